// CompactBilinearPoolingTSP_42262478193121
// MI455X (gfx1250) — compile-verified
//
#include <hip/hip_runtime.h>
#include <hip/hip_bf16.h>
#include <math.h>

// Problem constants (fixed by the reference)
#define B_   32
#define S_   145
#define C_   768
#define D_   8192
#define SN_  64
#define NK_  4097           // spectrum bins kept: k = 0..D/2 (conjugate symmetry)

// Workspace layout (units: floats). Total ~2.3 MB — sits in L2 (192 MB).
#define OFF_PROJ 0                   // [B, C]           sensor projection
#define OFF_H    24576               // [NK, 4]          H1,H2 interleaved
#define OFF_CV   40976               // [NK, 2]          Cv = H1*H2
#define AROW     8200                // padded complex row stride (>= 2*NK)
#define OFF_A    49176               // [B, AROW]        A  = G1*G2
#define OFF_BV   311576              // [B, AROW]        Bv = G1*H2 + H1*G2
#define OFF_IP   573976              // [B, S]           ip

typedef __attribute__((ext_vector_type(2)))  float        v2f;
typedef __attribute__((ext_vector_type(8)))  float        v8f;
typedef __attribute__((ext_vector_type(4)))  unsigned int v4u;
typedef __attribute__((ext_vector_type(8)))  int          v8i;
typedef __attribute__((ext_vector_type(4)))  int          v4i;

// ---------------------------------------------------------------------------
// Kernel A: proj[b,c] = sensor[b,:] . W_sensor[c,:] + b_sensor[c]
// 32x768 GEMM, K=64, via V_WMMA_F32_16X16X4_F32. One wave32 per 16x16 tile.
// ---------------------------------------------------------------------------
__global__ __launch_bounds__(32) void sensor_proj_wmma(
    const float* __restrict__ sensor, const float* __restrict__ Wsen,
    const float* __restrict__ bsen, float* __restrict__ proj) {
#if defined(__HIP_DEVICE_COMPILE__)
  const int lane = threadIdx.x;
  const int m0 = blockIdx.y * 16;
  const int n0 = blockIdx.x * 16;
  const int hi  = lane >> 4;     // 0: K pair {0,1}; 1: K pair {2,3} per ISA A/B layout
  const int l16 = lane & 15;
  v8f acc = {};
#pragma unroll
  for (int kk = 0; kk < SN_; kk += 4) {
    v2f a, b;
    // A (16x4): lanes 0-15 hold M=0..15 K={kk,kk+1}; lanes 16-31 K={kk+2,kk+3}
    a.x = sensor[(m0 + l16) * SN_ + kk + (hi ? 2 : 0)];
    a.y = sensor[(m0 + l16) * SN_ + kk + (hi ? 3 : 1)];
    // B (4x16): B[k][n] = W_sensor[n][k]; rows striped across lanes per VGPR
    b.x = Wsen[(n0 + l16) * SN_ + kk + (hi ? 2 : 0)];
    b.y = Wsen[(n0 + l16) * SN_ + kk + (hi ? 3 : 1)];
    acc = __builtin_amdgcn_wmma_f32_16x16x4_f32(false, a, false, b,
                                                (short)0, acc, false, false);
  }
  const int col = n0 + l16;
  const float bias = bsen[col];
#pragma unroll
  for (int r = 0; r < 8; ++r) {
    const int row = m0 + r + (hi ? 8 : 0);          // C/D layout: lanes16-31 -> M+8
    proj[row * C_ + col] = acc[r] + bias;
  }
#endif
}

// ---------------------------------------------------------------------------
// In-place radix-2 DIT FFT, N=8192, input pre-scattered in bit-reversed order,
// natural-order output. 256 threads, 16 butterflies/thread/stage, 13 stages.
// ---------------------------------------------------------------------------
__device__ __forceinline__ void fft8192(float* re, float* im, int tid) {
#pragma unroll 1
  for (int st = 0; st < 13; ++st) {
    const int half = 1 << st;
    const float angs = -3.14159265358979323846f / (float)half;
    for (int j = tid; j < (D_ >> 1); j += 256) {
      const int p  = j & (half - 1);
      const int g  = j >> st;
      const int i0 = (g << (st + 1)) + p;
      const int i1 = i0 + half;
      float sn, cs;
      __sincosf(angs * (float)p, &sn, &cs);
      const float ur = re[i0], ui = im[i0];
      const float vr = re[i1], vi = im[i1];
      const float tr = fmaf(cs, vr, -sn * vi);
      const float ti = fmaf(cs, vi,  sn * vr);
      re[i0] = ur + tr; im[i0] = ui + ti;
      re[i1] = ur - tr; im[i1] = ui - ti;
    }
    __syncthreads();
  }
}

// Unpack two real-signal spectra from one packed complex FFT:
// Z = FFT(q1 + i q2)  ->  F1[k], F2[k]
__device__ __forceinline__ void unpack2(const float* re, const float* im, int k,
                                        float& f1r, float& f1i,
                                        float& f2r, float& f2i) {
  const int m = (D_ - k) & (D_ - 1);
  const float zr = re[k], zi = im[k];
  const float wr = re[m], wi = im[m];
  f1r =  0.5f * (zr + wr);
  f1i =  0.5f * (zi - wi);
  f2r =  0.5f * (zi + wi);
  f2i = -0.5f * (zr - wr);
}

__device__ __forceinline__ int rev13(int x) { return (int)(__brev((unsigned)x) >> 19); }

// ---------------------------------------------------------------------------
// Kernel H: FFT of the "ones" sketches -> H1,H2 and Cv = H1*H2. One block.
// ---------------------------------------------------------------------------
__global__ __launch_bounds__(256) void ones_fft_kernel(
    const int* __restrict__ h1, const int* __restrict__ h2,
    const int* __restrict__ s1, const int* __restrict__ s2, float* ws) {
  __shared__ float re[D_];
  __shared__ float im[D_];
  const int tid = threadIdx.x;
  for (int i = tid; i < D_; i += 256) { re[i] = 0.f; im[i] = 0.f; }
  __syncthreads();
#pragma unroll
  for (int j = 0; j < 3; ++j) {
    const int c = tid + j * 256;
    atomicAdd(&re[rev13(h1[c])], (float)s1[c]);
    atomicAdd(&im[rev13(h2[c])], (float)s2[c]);
  }
  __syncthreads();
  fft8192(re, im, tid);
  float* Hb = ws + OFF_H;
  float* Cv = ws + OFF_CV;
  for (int k = tid; k <= D_ / 2; k += 256) {
    float h1r, h1i, h2r, h2i;
    unpack2(re, im, k, h1r, h1i, h2r, h2i);
    Hb[k * 4 + 0] = h1r; Hb[k * 4 + 1] = h1i;
    Hb[k * 4 + 2] = h2r; Hb[k * 4 + 3] = h2i;
    Cv[k * 2 + 0] = h1r * h2r - h1i * h2i;
    Cv[k * 2 + 1] = h1r * h2i + h1i * h2r;
  }
}

// ---------------------------------------------------------------------------
// Kernel Y: per-b sketches of proj, packed FFT -> G1,G2; emit A, Bv rows.
// ---------------------------------------------------------------------------
__global__ __launch_bounds__(256) void ybranch_kernel(
    const int* __restrict__ h1, const int* __restrict__ h2,
    const int* __restrict__ s1, const int* __restrict__ s2, float* ws) {
  __shared__ float re[D_];
  __shared__ float im[D_];
  const int b = blockIdx.x;
  const int tid = threadIdx.x;
  const float* proj = ws + OFF_PROJ + b * C_;
  for (int i = tid; i < D_; i += 256) { re[i] = 0.f; im[i] = 0.f; }
  __syncthreads();
#pragma unroll
  for (int j = 0; j < 3; ++j) {
    const int c = tid + j * 256;
    const float v = proj[c];
    atomicAdd(&re[rev13(h1[c])], v * (float)s1[c]);
    atomicAdd(&im[rev13(h2[c])], v * (float)s2[c]);
  }
  __syncthreads();
  fft8192(re, im, tid);
  const float* Hb = ws + OFF_H;
  float* Ab = ws + OFF_A  + b * AROW;
  float* Bb = ws + OFF_BV + b * AROW;
  for (int k = tid; k <= D_ / 2; k += 256) {
    float g1r, g1i, g2r, g2i;
    unpack2(re, im, k, g1r, g1i, g2r, g2i);
    const float h1r = Hb[k * 4], h1i = Hb[k * 4 + 1];
    const float h2r = Hb[k * 4 + 2], h2i = Hb[k * 4 + 3];
    Ab[k * 2 + 0] = g1r * g2r - g1i * g2i;
    Ab[k * 2 + 1] = g1r * g2i + g1i * g2r;
    Bb[k * 2 + 0] = (g1r * h2r - g1i * h2i) + (h1r * g2r - h1i * g2i);
    Bb[k * 2 + 1] = (g1r * h2i + g1i * h2r) + (h1r * g2i + h1i * g2r);
  }
}

// ---------------------------------------------------------------------------
// Main kernel: one workgroup per (b,s). TDM-load image row -> LDS, scatter
// packed sketch (bit-reversed), in-LDS FFT, Parseval reduction vs A/Bv/Cv.
// ip[b,s] = (w^2 dotA + w*beta dotB + beta^2 dotC) / D
// ---------------------------------------------------------------------------
__global__ __launch_bounds__(256) void tsp_main_kernel(
    const float* __restrict__ img, const float* __restrict__ tok1,
    const int* __restrict__ h1, const int* __restrict__ h2,
    const int* __restrict__ s1, const int* __restrict__ s2,
    const float* __restrict__ Ws2, const float* __restrict__ bs2,
    const float* __restrict__ ws_ro, float* __restrict__ ip_out) {
  __shared__ float re[D_];
  __shared__ float im[D_];
  const int s = blockIdx.x, b = blockIdx.y;
  const int tid = threadIdx.x;
  const float* row = img + ((size_t)b * S_ + s) * (size_t)C_;

  float v0, v1, v2;
#if defined(__HIP_DEVICE_COMPILE__) && defined(__gfx1250__) && \
    __has_builtin(__builtin_amdgcn_tensor_load_to_lds)
  // TDM: DMA the 768-float image row (1-D tile) into re[0..767].
  if (tid < 32) {                       // one wave issues the descriptor
    const unsigned long long ga = (unsigned long long)(size_t)row;
    const unsigned int lds = (unsigned int)(size_t)&re[0];
    v4u g0; v8i g1;
    v4i g2 = {0, 0, 0, 0};
    v4i g3 = {0, 0, 0, 0};
    v8i g4 = {0, 0, 0, 0, 0, 0, 0, 0};
    g0[0] = 1u;                                   // count=1, user descriptor
    g0[1] = lds;                                  // lds_addr
    g0[2] = (unsigned)(ga & 0xffffffffu);         // global_addr[31:0]
    g0[3] = (unsigned)((ga >> 32) & 0x1ffffffu) | (2u << 30); // addr[56:32], type=2
    g1[0] = 0x00020000;                           // wg_mask=0, data_size=2 (4B)
    g1[1] = (C_ & 0xffff) << 16;                  // tensor_dim0 = 768 (lo16)
    g1[2] = (1 << 16);                            // tensor_dim1 = 1
    g1[3] = (C_ << 16);                           // tile_dim0 = 768
    g1[4] = 1;                                    // tile_dim1 = 1, tile_dim2 = 0
    g1[5] = C_;                                   // tensor_dim0_stride = 768
    g1[6] = 0; g1[7] = 0;
    __builtin_amdgcn_tensor_load_to_lds(g0, g1, g2, g3, g4, 0);
    __builtin_amdgcn_s_wait_tensorcnt(0);
  }
  __syncthreads();
  v0 = re[tid]       + tok1[tid];
  v1 = re[tid + 256] + tok1[tid + 256];
  v2 = re[tid + 512] + tok1[tid + 512];
#else
  v0 = row[tid]       + tok1[tid];
  v1 = row[tid + 256] + tok1[tid + 256];
  v2 = row[tid + 512] + tok1[tid + 512];
#endif
  __syncthreads();
  for (int i = tid; i < D_; i += 256) { re[i] = 0.f; im[i] = 0.f; }
  __syncthreads();
  // Count-sketch scatter, directly into bit-reversed slots (feeds DIT FFT).
  {
    const int c0 = tid, c1 = tid + 256, c2 = tid + 512;
    atomicAdd(&re[rev13(h1[c0])], v0 * (float)s1[c0]);
    atomicAdd(&im[rev13(h2[c0])], v0 * (float)s2[c0]);
    atomicAdd(&re[rev13(h1[c1])], v1 * (float)s1[c1]);
    atomicAdd(&im[rev13(h2[c1])], v1 * (float)s2[c1]);
    atomicAdd(&re[rev13(h1[c2])], v2 * (float)s1[c2]);
    atomicAdd(&im[rev13(h2[c2])], v2 * (float)s2[c2]);
  }
  __syncthreads();
  fft8192(re, im, tid);

  // Parseval reduction over half-spectrum against per-b L2-resident vectors.
  const float* Ab = ws_ro + OFF_A  + b * AROW;
  const float* Bb = ws_ro + OFF_BV + b * AROW;
  const float* Cv = ws_ro + OFF_CV;
  float dA = 0.f, dB = 0.f, dC = 0.f;
  for (int k = tid; k <= D_ / 2; k += 256) {
    float f1r, f1i, f2r, f2i;
    unpack2(re, im, k, f1r, f1i, f2r, f2i);
    const float pr = f1r * f2r - f1i * f2i;     // Px = F1x * F2x
    const float pi = f1r * f2i + f1i * f2r;
    const float w = (k == 0 || k == D_ / 2) ? 1.f : 2.f;
    dA += w * (pr * Ab[2 * k] + pi * Ab[2 * k + 1]);
    dB += w * (pr * Bb[2 * k] + pi * Bb[2 * k + 1]);
    dC += w * (pr * Cv[2 * k] + pi * Cv[2 * k + 1]);
  }
  __syncthreads();
  re[tid] = dA; re[256 + tid] = dB; re[512 + tid] = dC;
  __syncthreads();
  for (int off = 128; off > 0; off >>= 1) {
    if (tid < off) {
      re[tid]       += re[tid + off];
      re[256 + tid] += re[256 + tid + off];
      re[512 + tid] += re[512 + tid + off];
    }
    __syncthreads();
  }
  if (tid == 0) {
    const float w = Ws2[s], bt = bs2[s];
    ip_out[b * S_ + s] =
        (w * w * re[0] + w * bt * re[256] + bt * bt * re[512]) * (1.0f / (float)D_);
  }
}

// ---------------------------------------------------------------------------
// Final: signed-sqrt, L2-normalize over S, project with W_out. One block / b.
// ---------------------------------------------------------------------------
__global__ __launch_bounds__(256) void final_kernel(
    const float* __restrict__ ip, const float* __restrict__ Wout,
    const float* __restrict__ bout, float* __restrict__ out) {
  __shared__ float sh[512];
  const int b = blockIdx.x, tid = threadIdx.x;
  float bp = 0.f, wd = 0.f;
  if (tid < S_) {
    const float v = ip[b * S_ + tid];
    const float sgn = (v > 0.f) ? 1.f : ((v < 0.f) ? -1.f : 0.f);
    bp = sgn * sqrtf(fabsf(v) + 1e-5f);
    wd = bp * Wout[tid];
  }
  sh[tid] = bp * bp; sh[256 + tid] = wd;
  __syncthreads();
  for (int off = 128; off > 0; off >>= 1) {
    if (tid < off) { sh[tid] += sh[tid + off]; sh[256 + tid] += sh[256 + tid + off]; }
    __syncthreads();
  }
  if (tid == 0) {
    const float n = fmaxf(sqrtf(sh[0]), 1e-12f);
    out[b] = sh[256] / n + bout[0];
  }
}

// ---------------------------------------------------------------------------
extern "C" void kernel_launch(void* const* d_in, const int* in_sizes, int n_in,
                              void* d_out, int out_size, void* d_ws, size_t ws_size,
                              hipStream_t stream) {
  (void)in_sizes; (void)n_in; (void)out_size; (void)ws_size;
  const float* sensor = (const float*)d_in[0];
  const float* img    = (const float*)d_in[1];
  // d_in[2] = image_masks (shape-only in reference), d_in[7] = d (fixed 8192)
  const int*   h1     = (const int*)d_in[3];
  const int*   h2     = (const int*)d_in[4];
  const int*   s1     = (const int*)d_in[5];
  const int*   s2     = (const int*)d_in[6];
  const float* Wsen   = (const float*)d_in[8];
  const float* bsen   = (const float*)d_in[9];
  const float* Ws2    = (const float*)d_in[10];
  const float* bs2    = (const float*)d_in[11];
  const float* Wout   = (const float*)d_in[12];
  const float* bout   = (const float*)d_in[13];
  const float* tok1   = (const float*)d_in[14] + C_;   // tok_emb row 1
  float* ws  = (float*)d_ws;
  float* out = (float*)d_out;

  sensor_proj_wmma<<<dim3(C_ / 16, B_ / 16), 32, 0, stream>>>(
      sensor, Wsen, bsen, ws + OFF_PROJ);
  ones_fft_kernel<<<1, 256, 0, stream>>>(h1, h2, s1, s2, ws);
  ybranch_kernel<<<B_, 256, 0, stream>>>(h1, h2, s1, s2, ws);
  tsp_main_kernel<<<dim3(S_, B_), 256, 0, stream>>>(
      img, tok1, h1, h2, s1, s2, Ws2, bs2, ws, ws + OFF_IP);
  final_kernel<<<B_, 256, 0, stream>>>(ws + OFF_IP, Wout, bout, out);
}